// WavetableSynth_36447092474141
// MI455X (gfx1250) — compile-verified
//
#include <hip/hip_runtime.h>
#include <math.h>

#define SR_F       44100.0f
#define NWT        64
#define WTLEN      513
#define LDS_STRIDE 65   // 65 ≡ 1 (mod 64 banks): row index spreads across banks

typedef __attribute__((ext_vector_type(2))) float v2f;
typedef __attribute__((ext_vector_type(8))) float v8f;

// ---------------- Phase 1: per-block partial sums of increment ----------------
__global__ __launch_bounds__(256) void k_block_sums(const float* __restrict__ pitch,
                                                    float* __restrict__ blockSums,
                                                    int T, int CT) {
  __shared__ float red[256];
  const float scale = (float)WTLEN / SR_F;
  int tid = threadIdx.x;
  long base = ((long)blockIdx.x * 256 + tid) * CT;
  float s = 0.f;
  for (int k = 0; k < CT; ++k) {
    long t = base + k;
    if (t < T) s += pitch[t] * scale;
  }
  red[tid] = s;
  __syncthreads();
  for (int off = 128; off > 0; off >>= 1) {
    if (tid < off) red[tid] += red[tid + off];
    __syncthreads();
  }
  if (tid == 0) blockSums[blockIdx.x] = red[0];
}

// ---------------- Phase 2: exclusive scan of block sums (1 block) -------------
__global__ __launch_bounds__(256) void k_scan_sums(float* __restrict__ blockSums, int nblk) {
  __shared__ float tmp[256];
  int i = threadIdx.x;
  float v = (i < nblk) ? blockSums[i] : 0.f;
  tmp[i] = v;
  __syncthreads();
  for (int off = 1; off < 256; off <<= 1) {
    float t = (i >= off) ? tmp[i - off] : 0.f;
    __syncthreads();
    tmp[i] += t;
    __syncthreads();
  }
  if (i < nblk) blockSums[i] = tmp[i] - v;   // exclusive prefix
}

// ---------------- Phase 3: write full phase-index array -----------------------
__global__ __launch_bounds__(256) void k_write_index(const float* __restrict__ pitch,
                                                     const float* __restrict__ blockOff,
                                                     float* __restrict__ index,
                                                     int T, int CT) {
  __shared__ float tsum[256];
  const float scale = (float)WTLEN / SR_F;
  int tid = threadIdx.x;
  long base = ((long)blockIdx.x * 256 + tid) * CT;
  float s = 0.f;
  for (int k = 0; k < CT; ++k) {
    long t = base + k;
    if (t < T) s += pitch[t] * scale;
  }
  tsum[tid] = s;
  __syncthreads();
  for (int off = 1; off < 256; off <<= 1) {
    float t = (tid >= off) ? tsum[tid - off] : 0.f;
    __syncthreads();
    tsum[tid] += t;
    __syncthreads();
  }
  float run  = blockOff[blockIdx.x] + (tsum[tid] - s);  // global exclusive prefix
  float inc0 = pitch[0] * scale;
  for (int k = 0; k < CT; ++k) {
    long t = base + k;
    if (t < T) {
      run += pitch[t] * scale;        // inclusive cumsum, matches jnp.cumsum f32
      index[t] = run - inc0;          // mod 513 applied later
    }
  }
}

// ---------------- Phase 4: fused interp + softmax + mix via WMMA --------------
// One wave handles a tile of 16 samples. Contraction over 64 wavetables done as
// D(16x16) = A(16x64) * B(64x16) via 16x V_WMMA_F32_16X16X4_F32; the wanted
// per-sample dot products are the diagonal of D.
__global__ __launch_bounds__(256) void k_synth(const float* __restrict__ amplitude,
                                               const float* __restrict__ wavetables,
                                               const float* __restrict__ attention,
                                               const float* __restrict__ index,
                                               float* __restrict__ out, int T) {
  __shared__ float wtsT[WTLEN * LDS_STRIDE];   // transposed, padded (131 KB < 320 KB)
  int tid = threadIdx.x;

  // Stage wavetable transposed into LDS; wts[:,512] = wavetables[:,0] (wrap).
  for (int i = tid; i < NWT * WTLEN; i += 256) {
    int w = i / WTLEN;
    int c = i - w * WTLEN;
    float v = wavetables[i];                    // coalesced read
    if (c < WTLEN - 1) wtsT[c * LDS_STRIDE + w] = v;
    if (c == 0)        wtsT[(WTLEN - 1) * LDS_STRIDE + w] = v;
  }
  __syncthreads();

  const int lane   = tid & 31;
  const int half   = lane >> 4;
  const int lane16 = lane & 15;
  const int kb     = half * 2;                  // K-pair base for this half-wave

  const int waveId = (((int)blockIdx.x * (int)blockDim.x) + tid) >> 5;
  const int nWaves = (((int)gridDim.x * (int)blockDim.x)) >> 5;
  const int nTiles = (T + 15) >> 4;

  for (int tile = waveId; tile < nTiles; tile += nWaves) {
    const int t  = tile * 16 + lane16;
    const int tc = (t < T) ? t : (T - 1);       // clamp: EXEC stays all-ones

    float idxf  = fmodf(index[tc], (float)WTLEN);
    float flo   = floorf(idxf);
    float alpha = idxf - flo;
    int lo = (int)flo;
    int hi = (int)ceilf(idxf);
    if (hi >= WTLEN) hi = 0;                    // ceil(index) % wt_len
    const int ldsLo = lo * LDS_STRIDE;
    const int ldsHi = hi * LDS_STRIDE;

    v8f  acc  = {0.f, 0.f, 0.f, 0.f, 0.f, 0.f, 0.f, 0.f};
    float ssum = 0.f;                           // per-lane softmax denom partial

    #pragma unroll
    for (int w0 = 0; w0 < NWT; w0 += 4) {
      const int w = w0 + kb;
      // A[m, w], A[m, w+1]: interpolated wavetable values for sample m = lane16
      float l0 = wtsT[ldsLo + w];
      float h0 = wtsT[ldsHi + w];
      float l1 = wtsT[ldsLo + w + 1];
      float h1 = wtsT[ldsHi + w + 1];
      v2f a;
      a.x = l0 + alpha * (h0 - l0);
      a.y = l1 + alpha * (h1 - l1);
      // B[w, n], B[w+1, n]: exp(attention) for sample n = lane16
      float e0 = __expf(attention[(size_t)w       * (size_t)T + (size_t)tc]);
      float e1 = __expf(attention[(size_t)(w + 1) * (size_t)T + (size_t)tc]);
      ssum += e0 + e1;
      v2f b; b.x = e0; b.y = e1;
      // D = A x B + C   (f32, 16x16x4)
      acc = __builtin_amdgcn_wmma_f32_16x16x4_f32(false, a, false, b,
                                                  (short)0, acc, false, false);
    }

    // softmax denominator: lane n and n+16 each summed half the 64 rows
    float S = ssum + __shfl_xor(ssum, 16, 32);

    // Diagonal of D: lanes 0-7 hold D[m,m] in element m; lanes 24-31 in elem lane-24
    const bool diag = (lane < 8) || (lane >= 24);
    const int  elem = (lane < 8) ? lane : ((lane >= 24) ? (lane - 24) : 0);
    const int  tOut = tile * 16 + lane16;
    if (diag && tOut < T) {
      float numer = acc[elem];
      out[tOut] = (numer / S) * amplitude[tOut];
    }
  }
}

// ------------------------------- launcher -------------------------------------
extern "C" void kernel_launch(void* const* d_in, const int* in_sizes, int n_in,
                              void* d_out, int out_size, void* d_ws, size_t ws_size,
                              hipStream_t stream) {
  const float* pitch      = (const float*)d_in[0];
  const float* amplitude  = (const float*)d_in[1];
  const float* wavetables = (const float*)d_in[2];
  const float* attention  = (const float*)d_in[3];
  const int T = in_sizes[0];

  float* index     = (float*)d_ws;     // T floats
  float* blockSums = index + T;        // <=256 floats

  // Choose per-thread chunk so the number of scan blocks is <= 256.
  int CT   = (T + 256 * 256 - 1) / (256 * 256);
  int nblk = (T + 256 * CT - 1) / (256 * CT);

  k_block_sums<<<nblk, 256, 0, stream>>>(pitch, blockSums, T, CT);
  k_scan_sums <<<1,    256, 0, stream>>>(blockSums, nblk);
  k_write_index<<<nblk, 256, 0, stream>>>(pitch, blockSums, index, T, CT);

  int nTiles = (T + 15) / 16;
  int blocks = (nTiles + 7) / 8;       // 8 waves per block, one tile per wave min
  if (blocks > 512) blocks = 512;      // keep LDS-staging traffic bounded
  if (blocks < 1)   blocks = 1;
  k_synth<<<blocks, 256, 0, stream>>>(amplitude, wavetables, attention, index,
                                      (float*)d_out, T);
}